// AttnResModule_28338194219040
// MI455X (gfx1250) — compile-verified
//
#include <hip/hip_runtime.h>
#include <math.h>

#define AS1 __attribute__((address_space(1)))
#define AS3 __attribute__((address_space(3)))
typedef float f4 __attribute__((ext_vector_type(4)));
typedef int   i4 __attribute__((ext_vector_type(4)));

constexpr int   D             = 2048;   // feature dim (reference-fixed)
constexpr int   NSTACK        = 4;      // softmax stack size n (reference-fixed)
constexpr int   THREADS       = 256;    // 8 waves of 32 on gfx1250
constexpr int   V4_PER_ROW    = D / 4;              // 512 float4 per n-row
constexpr int   V4_PER_THREAD = V4_PER_ROW / THREADS; // 2 float4 per thread per n-row
constexpr int   TOK_PER_BLOCK = 8;
constexpr float EPS           = 1.1920928955078125e-07f; // np.finfo(np.float32).eps

// ---- CDNA5 async global->LDS path (ASYNCcnt), with compile-safe fallback ----
#if defined(__has_builtin)
#if __has_builtin(__builtin_amdgcn_global_load_async_to_lds_b128)
#define USE_ASYNC 1
#endif
#endif
#ifndef USE_ASYNC
#define USE_ASYNC 0
#endif

__device__ __forceinline__ void lds_copy16(const float* g, float* l) {
#if USE_ASYNC
  // global_load_async_to_lds_b128: 16B/lane, bypasses VGPRs, tracked by ASYNCcnt
  __builtin_amdgcn_global_load_async_to_lds_b128((AS1 i4*)g, (AS3 i4*)l,
                                                 /*offset=*/0, /*cpol=*/0);
#else
  *(f4*)l = __builtin_nontemporal_load((const f4*)g);
#endif
}

template <int N>
__device__ __forceinline__ void wait_async() {
#if USE_ASYNC
#if __has_builtin(__builtin_amdgcn_s_wait_asynccnt)
  __builtin_amdgcn_s_wait_asynccnt(N);
#else
  asm volatile("s_wait_asynccnt %0" ::"n"(N) : "memory");
#endif
#endif
}

__global__ __launch_bounds__(THREADS) void attn_res_kernel(
    const float* __restrict__ V, const float* __restrict__ q,
    float* __restrict__ out, int ntok) {
  // Double-buffered token tiles: 2 x 4 x 2048 fp32 = 64 KB (WGP has 320 KB LDS)
  __shared__ float sV[2][NSTACK][D];
  __shared__ float sRed[THREADS / 32][2 * NSTACK];  // per-wave partials
  __shared__ float sW[NSTACK];                      // softmax weights

  const int tid  = threadIdx.x;
  const int wave = tid >> 5;
  const int lane = tid & 31;
  const long long rowStride = (long long)ntok * D;  // stride between n-slices of V

  // query is reused by every token: hold this thread's 8 q values in VGPRs
  f4 qr[V4_PER_THREAD];
#pragma unroll
  for (int j = 0; j < V4_PER_THREAD; ++j)
    qr[j] = ((const f4*)q)[tid + j * THREADS];

  const int tok0   = blockIdx.x * TOK_PER_BLOCK;
  const int tokEnd = (tok0 + TOK_PER_BLOCK < ntok) ? tok0 + TOK_PER_BLOCK : ntok;
  if (tok0 >= ntok) return;

  auto issue = [&](int tok, int buf) {
    const float* base = V + (long long)tok * D;
#pragma unroll
    for (int n = 0; n < NSTACK; ++n) {
      const float* gr = base + (long long)n * rowStride;
      float* lr = &sV[buf][n][0];
#pragma unroll
      for (int j = 0; j < V4_PER_THREAD; ++j) {
        const int c = (tid + j * THREADS) * 4;  // byte-coalesced 16B chunks
        lds_copy16(gr + c, lr + c);
      }
    }
  };

  issue(tok0, 0);  // prime the pipeline: 8 async b128 per wave outstanding

  for (int tok = tok0; tok < tokEnd; ++tok) {
    const int  buf  = (tok - tok0) & 1;
    const bool pref = (tok + 1 < tokEnd);
    if (pref) issue(tok + 1, buf ^ 1);  // overlap next token's HBM traffic

    // loads retire in order: <=8 outstanding means current tile is in LDS
    if (pref) wait_async<NSTACK * V4_PER_THREAD>();
    else      wait_async<0>();
    __syncthreads();

    // ---- single LDS read: keep values in registers for the combine ----
    f4 v[NSTACK][V4_PER_THREAD];
    float ss[NSTACK], dq[NSTACK];
#pragma unroll
    for (int n = 0; n < NSTACK; ++n) {
      ss[n] = 0.f; dq[n] = 0.f;
      const f4* row = (const f4*)&sV[buf][n][0];
#pragma unroll
      for (int j = 0; j < V4_PER_THREAD; ++j) {
        const f4 x = row[tid + j * THREADS];
        v[n][j] = x;
        ss[n] += x.x * x.x + x.y * x.y + x.z * x.z + x.w * x.w;
        dq[n] += x.x * qr[j].x + x.y * qr[j].y + x.z * qr[j].z + x.w * qr[j].w;
      }
    }

    // wave32 shuffle tree (deterministic), then 8-wave LDS combine
#pragma unroll
    for (int off = 16; off >= 1; off >>= 1) {
#pragma unroll
      for (int n = 0; n < NSTACK; ++n) {
        ss[n] += __shfl_xor(ss[n], off, 32);
        dq[n] += __shfl_xor(dq[n], off, 32);
      }
    }
    if (lane == 0) {
#pragma unroll
      for (int n = 0; n < NSTACK; ++n) {
        sRed[wave][n]          = ss[n];
        sRed[wave][NSTACK + n] = dq[n];
      }
    }
    __syncthreads();

    if (tid == 0) {
      float S[NSTACK], Q[NSTACK];
#pragma unroll
      for (int n = 0; n < NSTACK; ++n) {
        float a = 0.f, b = 0.f;
#pragma unroll
        for (int w = 0; w < THREADS / 32; ++w) { a += sRed[w][n]; b += sRed[w][NSTACK + n]; }
        S[n] = a; Q[n] = b;
      }
      float lg[NSTACK], m = -INFINITY;
#pragma unroll
      for (int n = 0; n < NSTACK; ++n) {
        lg[n] = Q[n] * rsqrtf(S[n] * (1.0f / (float)D) + EPS);
        m = fmaxf(m, lg[n]);
      }
      float e[NSTACK], sum = 0.f;
#pragma unroll
      for (int n = 0; n < NSTACK; ++n) { e[n] = expf(lg[n] - m); sum += e[n]; }
      const float inv = 1.0f / sum;
#pragma unroll
      for (int n = 0; n < NSTACK; ++n) sW[n] = e[n] * inv;
    }
    __syncthreads();

    // ---- weighted combine from registers; stream out non-temporally ----
    const float w0 = sW[0], w1 = sW[1], w2 = sW[2], w3 = sW[3];
    f4* o = (f4*)(out + (long long)tok * D);
#pragma unroll
    for (int j = 0; j < V4_PER_THREAD; ++j) {
      const f4 r = w0 * v[0][j] + w1 * v[1][j] + w2 * v[2][j] + w3 * v[3][j];
      __builtin_nontemporal_store(r, &o[tid + j * THREADS]);
    }
    __syncthreads();  // buffer 'buf' safe to refill next iteration; sW safe to overwrite
  }
}

extern "C" void kernel_launch(void* const* d_in, const int* in_sizes, int n_in,
                              void* d_out, int out_size, void* d_ws, size_t ws_size,
                              hipStream_t stream) {
  const float* V = (const float*)d_in[0];  // [n,B,T,D] fp32
  const float* q = (const float*)d_in[1];  // [D] fp32
  float* out = (float*)d_out;              // [B,T,D] fp32

  const int d    = in_sizes[1];                       // 2048
  const int ntok = in_sizes[0] / (NSTACK * d);        // B*T = 16384
  const int blocks = (ntok + TOK_PER_BLOCK - 1) / TOK_PER_BLOCK;
  attn_res_kernel<<<blocks, THREADS, 0, stream>>>(V, q, out, ntok);
}